// GCN_20899310862661
// MI455X (gfx1250) — compile-verified
//
#include <hip/hip_runtime.h>
#include <hip/hip_bf16.h>

typedef __attribute__((ext_vector_type(2))) float v2f;
typedef __attribute__((ext_vector_type(8))) float v8f;

#define KDIM 128
#define KPAIRS 64          // KDIM/2, K-pair interleaved LDS layout
#define LDSW_STRIDE 65     // in float2 units: 520B row stride -> lane halves on disjoint banks

// ---------------------------------------------------------------- utilities

__global__ void zero_kernel(float* __restrict__ p, long long n) {
  long long i = (long long)blockIdx.x * blockDim.x + threadIdx.x;
  long long stride = (long long)gridDim.x * blockDim.x;
  for (; i < n; i += stride) p[i] = 0.0f;
}

__global__ void degree_kernel(const int* __restrict__ src, const int* __restrict__ dst,
                              float* __restrict__ deg_src, float* __restrict__ deg_dst, int E) {
  int e = blockIdx.x * blockDim.x + threadIdx.x;
  if (e < E) {
    atomicAdd(&deg_src[src[e]], 1.0f);
    atomicAdd(&deg_dst[dst[e]], 1.0f);
  }
}

__global__ void rsqrt_kernel(float* __restrict__ d, int n) {
  int i = blockIdx.x * blockDim.x + threadIdx.x;
  if (i < n) d[i] = rsqrtf(fmaxf(d[i], 1.0f));
}

// ------------------------------------------------------------- SpMM scatter
// agg[sidx[e], :] += x[gidx[e], :] * rg[gidx[e]]
// 32 lanes per edge, 4 floats per lane. Node features ~51 MB -> resident in
// the 192 MB L2, so the random gather + float atomics stay on-chip.
__global__ void scatter_kernel(const float* __restrict__ x, const float* __restrict__ rg,
                               const int* __restrict__ gidx, const int* __restrict__ sidx,
                               float* __restrict__ agg, int E) {
  long long t = (long long)blockIdx.x * blockDim.x + threadIdx.x;
  int e = (int)(t >> 5);
  if (e >= E) return;
  int c = ((int)t & 31) * 4;
  int g = gidx[e];
  int s = sidx[e];
  float scale = rg[g];
  float4 v = *(const float4*)(x + (size_t)g * KDIM + c);
  float* out = agg + (size_t)s * KDIM + c;
  atomicAdd(out + 0, v.x * scale);
  atomicAdd(out + 1, v.y * scale);
  atomicAdd(out + 2, v.z * scale);
  atomicAdd(out + 3, v.w * scale);
}

// -------------------------------------------------------------- WMMA GEMM
// out[M x Nout] = epi( ((A .* sA_rows) @ W) + bias , Cin )
//   mode 0: out = P
//   mode 1: out = P + Cin          (linear-branch accumulation)
//   mode 2: out = max(Cin, P)      (h = max(h, hb) fusion)
// A is M x 128 (ld=128), W is 128 x Nout, out/Cin ld = Nout.
// Block = 256 threads (8 waves); each wave computes a 16x64 strip with
// V_WMMA_F32_16X16X4_F32 (full fp32 to match the reference).
// W staged in LDS K-pair interleaved: lds[kp][n] = {W[2kp][n], W[2kp+1][n]}
// so every B fragment is one ds_load_b64 straight into a register pair.
__global__ __launch_bounds__(256)
void gemm_wmma_kernel(const float* __restrict__ A, const float* __restrict__ sA,
                      const float* __restrict__ W, int M, int Nout,
                      const float* __restrict__ bias, const float* __restrict__ Cin,
                      int mode, float* __restrict__ out) {
  __shared__ v2f ldsW[KPAIRS * LDSW_STRIDE];

  const int n0 = blockIdx.y * 64;
  const int rowBlock = blockIdx.x * 128;

  // stage 128x64 tile of W, K-pair interleaved (zero-pad columns >= Nout)
  for (int idx = threadIdx.x; idx < KPAIRS * 64; idx += 256) {
    int kp = idx >> 6;
    int n = idx & 63;
    int col = n0 + n;
    v2f w;
    w.x = (col < Nout) ? W[(size_t)(2 * kp)     * Nout + col] : 0.0f;
    w.y = (col < Nout) ? W[(size_t)(2 * kp + 1) * Nout + col] : 0.0f;
    ldsW[kp * LDSW_STRIDE + n] = w;
  }
  __syncthreads();

  const int wave = threadIdx.x >> 5;
  const int lane = threadIdx.x & 31;
  const int half = lane >> 4;
  const int l15 = lane & 15;

  // A fragment (16x4 f32 layout): lane row = l15, K-pair = kk/2 + half.
  const int m = rowBlock + wave * 16 + l15;
  const bool mval = (m < M);
  // Fold the row guard into the scale: ghost rows contribute exactly 0,
  // and the A load becomes an unconditional float2 load (no exec branches).
  const float srow = !mval ? 0.0f : (sA != nullptr ? sA[m] : 1.0f);
  const float* arow = A + (size_t)(mval ? m : 0) * KDIM;

  v8f acc[4];
  acc[0] = (v8f){}; acc[1] = (v8f){}; acc[2] = (v8f){}; acc[3] = (v8f){};

#pragma unroll 2
  for (int kk = 0; kk < KDIM; kk += 4) {
    const int kp = (kk >> 1) + half;               // this lane's K-pair index
    v2f a = *(const v2f*)(arow + 2 * kp);          // 8B-aligned unconditional load
    a.x *= srow;
    a.y *= srow;
    const v2f* brow = &ldsW[kp * LDSW_STRIDE + l15];
#pragma unroll
    for (int j = 0; j < 4; ++j) {
      v2f b = brow[j * 16];                        // single ds_load_b64
      acc[j] = __builtin_amdgcn_wmma_f32_16x16x4_f32(
          /*neg_a=*/false, a, /*neg_b=*/false, b,
          /*c_mod=*/(short)0, acc[j], /*reuse_a=*/false, /*reuse_b=*/false);
    }
  }

  // Epilogue. D layout: VGPR v, lane L -> row = v + 8*(L>>4), col = L&15.
#pragma unroll
  for (int j = 0; j < 4; ++j) {
    const int c = n0 + j * 16 + l15;
    if (c >= Nout) continue;
    const float bv = (bias != nullptr) ? bias[c] : 0.0f;
#pragma unroll
    for (int v = 0; v < 8; ++v) {
      const int r = rowBlock + wave * 16 + v + 8 * half;
      if (r >= M) continue;
      float val = acc[j][v] + bv;
      if (mode == 1)      val += Cin[(size_t)r * Nout + c];
      else if (mode == 2) val = fmaxf(Cin[(size_t)r * Nout + c], val);
      out[(size_t)r * Nout + c] = val;
    }
  }
}

// -------------------------------------------------------------- batch norm

__global__ void bn_reduce_kernel(const float* __restrict__ h, int M, float* __restrict__ bnbuf) {
  const int col = threadIdx.x;  // 128 threads = 128 columns (coalesced rows)
  int r0 = blockIdx.x * 256;
  int r1 = r0 + 256; if (r1 > M) r1 = M;
  float s = 0.0f, sq = 0.0f;
  for (int r = r0; r < r1; ++r) {
    float v = h[(size_t)r * KDIM + col];
    s += v;
    sq += v * v;
  }
  atomicAdd(&bnbuf[col], s);
  atomicAdd(&bnbuf[KDIM + col], sq);
}

__global__ void bn_norm_relu_kernel(float* __restrict__ h, int M,
                                    const float* __restrict__ bnbuf,
                                    const float* __restrict__ gamma,
                                    const float* __restrict__ beta) {
  long long i = (long long)blockIdx.x * blockDim.x + threadIdx.x;
  if (i >= (long long)M * KDIM) return;
  int col = (int)(i & (KDIM - 1));
  float invM = 1.0f / (float)M;
  float mu = bnbuf[col] * invM;
  float var = bnbuf[KDIM + col] * invM - mu * mu;
  float v = (h[i] - mu) * rsqrtf(var + 1e-5f) * gamma[col] + beta[col];
  h[i] = fmaxf(v, 0.0f);
}

// ---------------------------------------------------------------- launcher

static inline unsigned cdiv_u(long long a, long long b) { return (unsigned)((a + b - 1) / b); }

extern "C" void kernel_launch(void* const* d_in, const int* in_sizes, int n_in,
                              void* d_out, int out_size, void* d_ws, size_t ws_size,
                              hipStream_t stream) {
  const float* feat  = (const float*)d_in[0];   // N x 128
  const float* Wb    = (const float*)d_in[1];   // 3 x 128 x 128
  const float* bb2   = (const float*)d_in[2];   // 128
  const float* Wh01  = (const float*)d_in[3];   // 2 x 128 x 128
  const float* W2    = (const float*)d_in[4];   // 128 x 40
  const float* b2    = (const float*)d_in[5];   // 40
  const float* Wl01  = (const float*)d_in[6];   // 2 x 128 x 128
  const float* Wl2   = (const float*)d_in[7];   // 128 x 40
  const float* gamma = (const float*)d_in[8];   // 2 x 128
  const float* beta  = (const float*)d_in[9];   // 2 x 128
  const int*   src   = (const int*)d_in[10];    // E
  const int*   dst   = (const int*)d_in[11];    // E

  const int N  = in_sizes[0] / KDIM;
  const int E  = in_sizes[10];
  const int NC = in_sizes[5];                   // 40 classes

  // workspace layout (floats)
  float* ws    = (float*)d_ws;
  float* r_src = ws;                 ws += N;                // rsqrt(out-deg)
  float* r_dst = ws;                 ws += N;                // rsqrt(in-deg)
  float* agg   = ws;                 ws += (size_t)N * KDIM;
  float* B1    = ws;                 ws += (size_t)N * KDIM; // hmax
  float* B2    = ws;                 ws += (size_t)N * KDIM; // h (ping)
  float* bnbuf = ws;                 ws += 2 * KDIM;
  (void)ws_size; (void)n_in; (void)out_size;

  const dim3 blk(256);

  // --- degrees -> rsqrt norms (r_src,r_dst contiguous: one zero/rsqrt pass)
  zero_kernel<<<512, blk, 0, stream>>>(r_src, (long long)2 * N);
  degree_kernel<<<cdiv_u(E, 256), blk, 0, stream>>>(src, dst, r_src, r_dst, E);
  rsqrt_kernel<<<cdiv_u((long long)2 * N, 256), blk, 0, stream>>>(r_src, 2 * N);

  const unsigned gemm_gx = cdiv_u(N, 128);
  const unsigned scat_gx = cdiv_u((long long)E * 32, 256);

  const float* hcur = feat;
  for (int i = 0; i < 3; ++i) {
    const bool  last = (i == 2);
    const int   Nout = last ? NC : KDIM;
    const float* Wbi = Wb + (size_t)i * KDIM * KDIM;
    const float* Wi  = last ? W2  : Wh01 + (size_t)i * KDIM * KDIM;
    const float* Wli = last ? Wl2 : Wl01 + (size_t)i * KDIM * KDIM;
    float* hmax  = B1;
    float* hnext = last ? (float*)d_out : B2;

    // ---- backward GraphConv on reversed graph (applied to raw feat) ----
    // aggB[src[e]] += feat[dst[e]] * r_dst[dst[e]]
    zero_kernel<<<2048, blk, 0, stream>>>(agg, (long long)N * KDIM);
    scatter_kernel<<<scat_gx, blk, 0, stream>>>(feat, r_dst, dst, src, agg, E);
    // hmax = max(hcur, (aggB .* r_src) @ Wb[i] + (last ? bb2 : 0))
    gemm_wmma_kernel<<<dim3(gemm_gx, 2), blk, 0, stream>>>(
        agg, r_src, Wbi, N, KDIM, last ? bb2 : nullptr, hcur, /*mode=*/2, hmax);

    // ---- forward GraphConv + linear branch ----
    // aggF[dst[e]] += hmax[src[e]] * r_src[src[e]]
    zero_kernel<<<2048, blk, 0, stream>>>(agg, (long long)N * KDIM);
    scatter_kernel<<<scat_gx, blk, 0, stream>>>(hmax, r_src, src, dst, agg, E);
    // hnext = (aggF .* r_dst) @ W + (last ? b2 : 0)
    gemm_wmma_kernel<<<dim3(gemm_gx, cdiv_u(Nout, 64)), blk, 0, stream>>>(
        agg, r_dst, Wi, N, Nout, last ? b2 : nullptr, nullptr, /*mode=*/0, hnext);
    // hnext += hmax @ Wl
    gemm_wmma_kernel<<<dim3(gemm_gx, cdiv_u(Nout, 64)), blk, 0, stream>>>(
        hmax, nullptr, Wli, N, Nout, nullptr, hnext, /*mode=*/1, hnext);

    // ---- batch norm (biased var) + ReLU ----
    if (!last) {
      zero_kernel<<<1, blk, 0, stream>>>(bnbuf, 2 * KDIM);
      bn_reduce_kernel<<<cdiv_u(N, 256), dim3(KDIM), 0, stream>>>(hnext, N, bnbuf);
      bn_norm_relu_kernel<<<cdiv_u((long long)N * KDIM, 256), blk, 0, stream>>>(
          hnext, N, bnbuf, gamma + (size_t)i * KDIM, beta + (size_t)i * KDIM);
      hcur = hnext;  // B2 becomes next layer's input
    }
  }
}